// CODI_MiniBatch_Loss_75273596830476
// MI455X (gfx1250) — compile-verified
//
#include <hip/hip_runtime.h>
#include <hip/hip_bf16.h>

// CODI minibatch loss for MI455X (gfx1250).
//
// Single streaming pass over z (168 MB -> ~7.2us floor at 23.3 TB/s) using
//   sse_k = Q_k - ||S_k||^2 / n_k + n_k * eps^2 * CH
// Segment sums S and squared sums Q are computed as one-hot GEMMs with
// V_WMMA_F32_16X16X4_F32 (exact f32 accumulation; one-hot A has 0/1 entries).

typedef float v2f __attribute__((ext_vector_type(2)));
typedef float v8f __attribute__((ext_vector_type(8)));

#define B_TOTAL    8192
#define C_DIM      20
#define H_DIM      256
#define CH         (C_DIM * H_DIM)      // 5120 contiguous floats per sample
#define NUM_LABELS 10
#define K_PAD      16                   // WMMA M dimension (labels padded to 16)
#define NTILE      (CH / 16)            // 320 column tiles
#define EPS_F      1e-8f

__device__ __forceinline__ float fixnum(float v) {
  // jnp.nan_to_num: NaN -> 0, +/-inf -> +/-FLT_MAX
  v = (v != v) ? 0.0f : v;
  v = fminf(fmaxf(v, -3.402823466e+38f), 3.402823466e+38f);
  return v;
}

// ---------------------------------------------------------------------------
// Pass 1: one wave per (column-tile, sample-chunk). Accumulates
//   accS[m][n] += onehot(label==m) * z       accQ[m][n] += onehot(label==m) * z^2
// via V_WMMA_F32_16X16X4_F32; 16 samples (8 loads, 8 WMMAs) per loop trip.
//
// VGPR layouts (ISA 7.12.2, f32):
//   A 16x4 : lane = M (lane&15); {V0,V1} = K {0,1} (lanes 0-15) / {2,3} (16-31)
//   B 4x16 : lane = N (lane&15); {V0,V1} = K {0,1} (lanes 0-15) / {2,3} (16-31)
//   C/D    : lane = N (lane&15); VGPR r = M = r + (lane>=16 ? 8 : 0)
// ---------------------------------------------------------------------------
__global__ void __launch_bounds__(32)
codi_pass1(const float* __restrict__ z, const int* __restrict__ labels,
           float* __restrict__ partS, float* __restrict__ partQ,
           int chunk_len) {
  const int tile  = (int)blockIdx.x;
  const int chunk = (int)blockIdx.y;
  const int lane  = (int)threadIdx.x;   // 0..31, full wave, never diverges
  const int half  = lane >> 4;          // 0 | 1
  const int n     = lane & 15;
  const int c     = tile * 16 + n;      // global column this lane owns
  const int kb    = half * 2;           // K base for this lane (0 or 2)
  const int m     = n;                  // A-matrix row for this lane

  v8f accS = {0.f, 0.f, 0.f, 0.f, 0.f, 0.f, 0.f, 0.f};
  v8f accQ = {0.f, 0.f, 0.f, 0.f, 0.f, 0.f, 0.f, 0.f};

  const int b_beg = chunk * chunk_len;
  const int iters = chunk_len >> 4;     // 16 samples per trip

  // Single advancing base pointer; all loads/prefetches use imm24 offsets.
  const float* p  = z + (size_t)(b_beg + kb) * CH + c;
  const int4*  lp = (const int4*)(labels + b_beg);   // uniform -> s_load_b128

  for (int it = 0; it < iters; ++it) {
    // Prefetch next trip's 8 rows (~320 KB ahead), near-cache locality.
    if (it + 1 < iters) {
#pragma unroll
      for (int s = 0; s < 4; ++s) {
        __builtin_prefetch(p + (size_t)(16 + s * 4) * CH, 0, 3);
        __builtin_prefetch(p + (size_t)(17 + s * 4) * CH, 0, 3);
      }
    }

#pragma unroll
    for (int s = 0; s < 4; ++s) {
      const int4 lv = lp[s];                       // labels[b0+4s .. b0+4s+3]

      const float ax = ((half ? lv.z : lv.x) == m) ? 1.0f : 0.0f;
      const float ay = ((half ? lv.w : lv.y) == m) ? 1.0f : 0.0f;

      const float bx = fixnum(p[(size_t)(s * 4) * CH]);
      const float by = fixnum(p[(size_t)(s * 4 + 1) * CH]);

      const v2f A  = {ax, ay};
      const v2f Bm = {bx, by};
      const v2f B2 = {bx * bx, by * by};

      accS = __builtin_amdgcn_wmma_f32_16x16x4_f32(
          false, A, false, Bm, (short)0, accS, false, false);
      accQ = __builtin_amdgcn_wmma_f32_16x16x4_f32(
          false, A, false, B2, (short)0, accQ, false, false);
    }

    p  += (size_t)16 * CH;
    lp += 4;
  }

  // Deterministic write-out of per-chunk partials (no float atomics).
#pragma unroll
  for (int r = 0; r < 8; ++r) {
    const int mr = r + half * 8;
    partS[(size_t)(chunk * K_PAD + mr) * CH + c] = accS[r];
    partQ[(size_t)(chunk * K_PAD + mr) * CH + c] = accQ[r];
  }
}

// ---------------------------------------------------------------------------
// Pass 2: one block per label row m. Reduce chunk partials, produce
//   kstats[m] = { ||S_m||^2 , Q_m }
// ---------------------------------------------------------------------------
__global__ void __launch_bounds__(256)
codi_pass2(const float* __restrict__ partS, const float* __restrict__ partQ,
           float* __restrict__ kstats, int nchunk) {
  const int m = (int)blockIdx.x;
  const int t = (int)threadIdx.x;

  float s2 = 0.f, q = 0.f;
  for (int c = t; c < CH; c += 256) {
    float S = 0.f, Q = 0.f;
    for (int ch = 0; ch < nchunk; ++ch) {
      S += partS[(size_t)(ch * K_PAD + m) * CH + c];
      Q += partQ[(size_t)(ch * K_PAD + m) * CH + c];
    }
    s2 += S * S;
    q  += Q;
  }

  __shared__ float rs[256];
  __shared__ float rq[256];
  rs[t] = s2;
  rq[t] = q;
  __syncthreads();
  for (int s = 128; s > 0; s >>= 1) {
    if (t < s) { rs[t] += rs[t + s]; rq[t] += rq[t + s]; }
    __syncthreads();
  }
  if (t == 0) {
    kstats[m * 2 + 0] = rs[0];
    kstats[m * 2 + 1] = rq[0];
  }
}

// ---------------------------------------------------------------------------
// Pass 3: counts (integer LDS atomics -> deterministic) + final scalar.
// ---------------------------------------------------------------------------
__global__ void __launch_bounds__(256)
codi_pass3(const int* __restrict__ labels, const float* __restrict__ kstats,
           float* __restrict__ out) {
  __shared__ int cnt[K_PAD];
  const int t = (int)threadIdx.x;
  if (t < K_PAD) cnt[t] = 0;
  __syncthreads();
  for (int i = t; i < B_TOTAL; i += 256) {
    const int l = labels[i];
    if (l >= 0 && l < K_PAD) atomicAdd(&cnt[l], 1);
  }
  __syncthreads();
  if (t == 0) {
    float loss = 0.f;
    for (int k = 0; k < NUM_LABELS; ++k) {
      const float nk = (float)cnt[k];
      if (nk > 0.f) {
        const float S2  = kstats[k * 2 + 0];
        const float Q   = kstats[k * 2 + 1];
        const float sse = Q - S2 / nk + nk * (EPS_F * EPS_F) * (float)CH;
        loss += sse / (nk * (float)CH);
      }
    }
    out[0] = loss;
  }
}

// ---------------------------------------------------------------------------
extern "C" void kernel_launch(void* const* d_in, const int* in_sizes, int n_in,
                              void* d_out, int out_size, void* d_ws, size_t ws_size,
                              hipStream_t stream) {
  (void)in_sizes; (void)n_in; (void)out_size;

  const float* z      = (const float*)d_in[0];   // [8192, 20, 256] f32
  const int*   labels = (const int*)d_in[1];     // [8192] i32
  float*       out    = (float*)d_out;           // scalar f32

  // Choose chunk count from available scratch (deterministic: ws_size is
  // fixed across calls). Prefer 32 chunks (10240 waves, ~21 MB scratch).
  const size_t per_chunk = (size_t)2 * K_PAD * CH * sizeof(float); // S+Q row
  int nchunk = 32;
  while (nchunk > 1 && (size_t)nchunk * per_chunk + 256 > ws_size) nchunk >>= 1;
  const int chunk_len = B_TOTAL / nchunk;        // power of two, >= 256

  float* partS  = (float*)d_ws;
  float* partQ  = partS + (size_t)nchunk * K_PAD * CH;
  float* kstats = partQ + (size_t)nchunk * K_PAD * CH;

  codi_pass1<<<dim3(NTILE, nchunk), 32, 0, stream>>>(z, labels, partS, partQ,
                                                     chunk_len);
  codi_pass2<<<K_PAD, 256, 0, stream>>>(partS, partQ, kstats, nchunk);
  codi_pass3<<<1, 256, 0, stream>>>(labels, kstats, out);
}